// PrefixTunedAttention_67800353734981
// MI455X (gfx1250) — compile-verified
//
#include <hip/hip_runtime.h>
#include <math.h>
#include <stdint.h>

// ---------------------------------------------------------------------------
// PrefixTunedAttention on gfx1250 (MI455X), fp32 end-to-end via
// V_WMMA_F32_16X16X4_F32 (exact fp32 matmul on the matrix cores).
// Memory-bound problem (~140 GFLOP vs ~350MB @ 23.3TB/s) -> keep fp32.
// K/V tiles in attention staged through LDS by the Tensor Data Mover.
// ---------------------------------------------------------------------------

typedef float f32x2 __attribute__((ext_vector_type(2)));
typedef float f32x8 __attribute__((ext_vector_type(8)));
typedef unsigned int u32x4 __attribute__((ext_vector_type(4)));
typedef int i32x4 __attribute__((ext_vector_type(4)));
typedef int i32x8 __attribute__((ext_vector_type(8)));

#define DMODEL 1024
#define ND3    3072
#define NHEAD  16
#define HDIM   64
#define SEQ    2048
#define NBATCH 4
#define NPFX   20
#define KVLEN  2068   /* NPFX + SEQ */
#define KVPAD  2080   /* 130 * 16   */
#define NTOK   8192   /* NBATCH*SEQ */
#define LDSPAD 68     /* 64 + 4 dword row pad from TDM pad feature */

__device__ __forceinline__ f32x8 wmma4(f32x2 a, f32x2 b, f32x8 c) {
  // (neg_a, A, neg_b, B, c_mod, C, reuse_a, reuse_b)
  return __builtin_amdgcn_wmma_f32_16x16x4_f32(false, a, false, b, (short)0, c,
                                               false, false);
}

// ---------------------------------------------------------------------------
// TDM: DMA one 16x64 fp32 tile (row stride 64 elems) global -> LDS, padding
// each 64-dword row with 4 dwords (LDS row stride = 68 floats, bank-conflict
// free column access).  D# layout per CDNA5 ISA.
// ---------------------------------------------------------------------------
__device__ __forceinline__ void tdm_load_tile16x64(float* lds_dst,
                                                   const float* gsrc) {
  const uint32_t lds = (uint32_t)(uintptr_t)lds_dst;   // LDS aperture: low 32b
  const uint64_t ga = (uint64_t)(uintptr_t)gsrc;
  u32x4 g0;
  g0[0] = 1u;                                          // count=1, user mode
  g0[1] = lds;                                         // lds_addr
  g0[2] = (uint32_t)ga;                                // global_addr[31:0]
  g0[3] = (uint32_t)((ga >> 32) & 0x01FFFFFFu) | 0x80000000u;  // addr | type=2
  i32x8 g1;
  // data_size=4B (2<<16) | pad_enable (1<<20) | pad_interval=64dw (5<<22)
  // | pad_amount=4dw (3<<25)
  g1[0] = (int)((2u << 16) | (1u << 20) | (5u << 22) | (3u << 25));
  g1[1] = (int)(64u << 16);   // tensor_dim0 = 64 (low 16 bits @ bit48)
  g1[2] = (int)(16u << 16);   // tensor_dim0 hi=0 | tensor_dim1 = 16
  g1[3] = (int)(64u << 16);   // tensor_dim1 hi=0 | tile_dim0 = 64
  g1[4] = 16;                 // tile_dim1 = 16 | tile_dim2 = 0
  g1[5] = 64;                 // tensor_dim0_stride = 64 elems
  g1[6] = 0;
  g1[7] = 0;
  i32x4 gz;
  gz[0] = 0; gz[1] = 0; gz[2] = 0; gz[3] = 0;
#if __clang_major__ >= 23
  i32x8 gz8;
  gz8[0] = 0; gz8[1] = 0; gz8[2] = 0; gz8[3] = 0;
  gz8[4] = 0; gz8[5] = 0; gz8[6] = 0; gz8[7] = 0;
  __builtin_amdgcn_tensor_load_to_lds(g0, g1, gz, gz, gz8, 0);
#else
  __builtin_amdgcn_tensor_load_to_lds(g0, g1, gz, gz, 0);
#endif
}

// ---------------------------------------------------------------------------
// LayerNorm: one wave per row of 1024, float4 I/O, shfl-xor reduction.
// ---------------------------------------------------------------------------
__global__ __launch_bounds__(256) void ln_kernel(const float* __restrict__ x,
                                                 const float* __restrict__ g,
                                                 const float* __restrict__ bt,
                                                 float* __restrict__ xn) {
  const int lane = threadIdx.x & 31;
  const int row  = blockIdx.x * 8 + (threadIdx.x >> 5);
  const float4* xr = (const float4*)(x + (size_t)row * DMODEL);
  float4 v[8];
  float s = 0.f, ss = 0.f;
#pragma unroll
  for (int i = 0; i < 8; ++i) {
    v[i] = xr[lane + i * 32];
    s  += v[i].x + v[i].y + v[i].z + v[i].w;
    ss += v[i].x * v[i].x + v[i].y * v[i].y + v[i].z * v[i].z + v[i].w * v[i].w;
  }
#pragma unroll
  for (int m = 1; m <= 16; m <<= 1) {
    s  += __shfl_xor(s,  m, 32);
    ss += __shfl_xor(ss, m, 32);
  }
  const float mu = s * (1.f / DMODEL);
  const float rs = rsqrtf(ss * (1.f / DMODEL) - mu * mu + 1e-5f);
  float4* xo = (float4*)(xn + (size_t)row * DMODEL);
  const float4* gp = (const float4*)g;
  const float4* bp = (const float4*)bt;
#pragma unroll
  for (int i = 0; i < 8; ++i) {
    const int c = lane + i * 32;
    const float4 gv = gp[c], bv = bp[c];
    float4 o;
    o.x = (v[i].x - mu) * rs * gv.x + bv.x;
    o.y = (v[i].y - mu) * rs * gv.y + bv.y;
    o.z = (v[i].z - mu) * rs * gv.z + bv.z;
    o.w = (v[i].w - mu) * rs * gv.w + bv.w;
    xo[c] = o;
  }
}

// ---------------------------------------------------------------------------
// Tiny prefix-MLP layers (20 rows): thread per output element. Negligible.
// ---------------------------------------------------------------------------
__global__ void mlp_kernel(const float* __restrict__ in,
                           const float* __restrict__ w,
                           const float* __restrict__ bias,
                           float* __restrict__ out,
                           int R, int K, int N, int do_tanh) {
  const int idx = blockIdx.x * blockDim.x + threadIdx.x;
  if (idx >= R * N) return;
  const int r = idx / N, j = idx % N;
  const float* ir = in + (size_t)r * K;
  float acc = bias[j];
  for (int k = 0; k < K; ++k) acc = fmaf(ir[k], w[(size_t)k * N + j], acc);
  out[idx] = do_tanh ? tanhf(acc) : acc;
}

// Broadcast prefix k/v into per-(b,h) kv buffers; zero the 12 pad rows.
__global__ void prefix_fill_kernel(const float* __restrict__ pk,
                                   const float* __restrict__ pv,
                                   float* __restrict__ kbuf,
                                   float* __restrict__ vbuf) {
  const int idx = blockIdx.x * blockDim.x + threadIdx.x;  // 4*16*32*64 = 131072
  if (idx >= NBATCH * NHEAD * 32 * HDIM) return;
  const int d  = idx & 63;
  const int rr = (idx >> 6) & 31;
  const int h  = (idx >> 11) & 15;
  const int bq = idx >> 15;
  const size_t base = (size_t)(bq * NHEAD + h) * KVPAD * HDIM;
  if (rr < NPFX) {
    kbuf[base + (size_t)rr * HDIM + d] = pk[rr * DMODEL + h * HDIM + d];
    vbuf[base + (size_t)rr * HDIM + d] = pv[rr * DMODEL + h * HDIM + d];
  } else {
    const int row = KVLEN + (rr - NPFX);
    kbuf[base + (size_t)row * HDIM + d] = 0.f;
    vbuf[base + (size_t)row * HDIM + d] = 0.f;
  }
}

// ---------------------------------------------------------------------------
// Wave-level 32x64 fp32-WMMA GEMM core: 8 x V_WMMA_F32_16X16X4_F32 per K=4.
// A-frag: lane m=lane&15, k = 2*(lane>>4)+{0,1};  B-frag: lane n=lane&15.
// ---------------------------------------------------------------------------
__device__ __forceinline__ void gemm32x64(const float* __restrict__ A, int lda,
                                          const float* __restrict__ B, int ldb,
                                          int K, int lane, f32x8 acc[2][4]) {
  const int hi = lane >> 4, mr = lane & 15;
  const float* a0p = A + (size_t)mr * lda;
  const float* a1p = A + (size_t)(mr + 16) * lda;
  for (int k = 0; k < K; k += 4) {
    const int kb = k + hi * 2;
    f32x2 a0 = {a0p[kb], a0p[kb + 1]};
    f32x2 a1 = {a1p[kb], a1p[kb + 1]};
    const float* bp = B + (size_t)kb * ldb + mr;
    __builtin_prefetch(bp + 4 * (size_t)ldb, 0, 3);  // WGP-scope prefetch
#pragma unroll
    for (int t = 0; t < 4; ++t) {
      f32x2 bf = {bp[t * 16], bp[t * 16 + ldb]};
      acc[0][t] = wmma4(a0, bf, acc[0][t]);
      acc[1][t] = wmma4(a1, bf, acc[1][t]);
    }
  }
}

// QKV projection: [8192,1024] @ [1024,3072]; scatter into q / k / v (k,v at
// row offset NPFX so the prefix concat is free).
__global__ __launch_bounds__(256) void qkv_gemm_kernel(
    const float* __restrict__ xn, const float* __restrict__ w,
    float* __restrict__ q, float* __restrict__ kbuf, float* __restrict__ vbuf) {
  const int lane = threadIdx.x & 31;
  const int W = __builtin_amdgcn_readfirstlane(blockIdx.x * 8 +
                                               (threadIdx.x >> 5));
  const int TN = ND3 / 64;  // 48
  const int m0 = (W / TN) * 32, n0 = (W % TN) * 64;
  const int hi = lane >> 4, mr = lane & 15;

  f32x8 acc[2][4];
#pragma unroll
  for (int i = 0; i < 2; ++i)
#pragma unroll
    for (int t = 0; t < 4; ++t)
#pragma unroll
      for (int r = 0; r < 8; ++r) acc[i][t][r] = 0.f;

  gemm32x64(xn + (size_t)m0 * DMODEL, DMODEL, w + n0, ND3, DMODEL, lane, acc);

#pragma unroll
  for (int mi = 0; mi < 2; ++mi)
#pragma unroll
    for (int t = 0; t < 4; ++t) {
      // col block coordinates are wave-uniform (scalar branches, no exec mask)
      const int col0 = n0 + t * 16;
      const int p = col0 >> 10;        // 0:q 1:k 2:v
      const int h = (col0 >> 6) & 15;
      const int d = (col0 & 63) + mr;
#pragma unroll
      for (int r = 0; r < 8; ++r) {
        const int tok = m0 + mi * 16 + r + 8 * hi;
        const int bq = tok >> 11, l = tok & 2047;
        const size_t bh = (size_t)(bq * NHEAD + h);
        const float val = acc[mi][t][r];
        if (p == 0)
          q[(bh * SEQ + l) * HDIM + d] = val;
        else if (p == 1)
          kbuf[(bh * KVPAD + NPFX + l) * HDIM + d] = val;
        else
          vbuf[(bh * KVPAD + NPFX + l) * HDIM + d] = val;
      }
    }
}

// ---------------------------------------------------------------------------
// Flash attention: 4 waves/block share one (b,h); K/V 16x64 tiles are DMA'd
// into LDS by the Tensor Data Mover (double buffered, TENSORcnt + barrier),
// then both Q@K^T and P@V run on fp32 WMMA.  Online softmax uses 16-lane
// xor-shuffle row reductions; P tile goes C-layout -> A-layout through LDS.
// ---------------------------------------------------------------------------
__global__ __launch_bounds__(128) void attn_kernel(
    const float* __restrict__ q, const float* __restrict__ kbuf,
    const float* __restrict__ vbuf, float* __restrict__ o) {
  __shared__ float Ks[2][16][LDSPAD];
  __shared__ float Vs[2][16][LDSPAD];
  __shared__ float Ps[4][16][17];
  const int lane = threadIdx.x & 31;
  const int wv = threadIdx.x >> 5;
  const int hi = lane >> 4, mr = lane & 15;
  const int q0 = __builtin_amdgcn_readfirstlane((blockIdx.x * 4 + wv) * 16);
  const int h = blockIdx.y, bq = blockIdx.z;
  const size_t bh = (size_t)(bq * NHEAD + h);
  const float* Q = q + (bh * SEQ + q0) * HDIM;
  const float* K = kbuf + bh * KVPAD * HDIM;
  const float* V = vbuf + bh * KVPAD * HDIM;

  f32x2 qf[16];
#pragma unroll
  for (int f = 0; f < 16; ++f) {
    const int kb = f * 4 + hi * 2;
    qf[f][0] = Q[mr * HDIM + kb];
    qf[f][1] = Q[mr * HDIM + kb + 1];
  }

  f32x8 oacc[4];
#pragma unroll
  for (int t = 0; t < 4; ++t)
#pragma unroll
    for (int r = 0; r < 8; ++r) oacc[t][r] = 0.f;
  float mrow[8], srow[8];
#pragma unroll
  for (int r = 0; r < 8; ++r) { mrow[r] = -1e30f; srow[r] = 0.f; }

  const float SCALE = 0.125f;  // 1/sqrt(64)
  const int NT = KVPAD / 16;   // 130

  // prologue: DMA tile 0 into buffer 0 (wave 0 owns the TDM)
  if (wv == 0) {
    tdm_load_tile16x64(&Ks[0][0][0], K);
    tdm_load_tile16x64(&Vs[0][0][0], V);
  }

  for (int it = 0; it < NT; ++it) {
    const int cur = it & 1;
    const int kv0 = it * 16;
    if (wv == 0) {
      if (it + 1 < NT) {
        // issue next tile's DMA, then wait for the current tile's pair
        tdm_load_tile16x64(&Ks[cur ^ 1][0][0], K + (size_t)(kv0 + 16) * HDIM);
        tdm_load_tile16x64(&Vs[cur ^ 1][0][0], V + (size_t)(kv0 + 16) * HDIM);
        __builtin_amdgcn_s_wait_tensorcnt(2);
      } else {
        __builtin_amdgcn_s_wait_tensorcnt(0);
      }
    }
    __syncthreads();  // current K/V tile visible to all 4 waves

    // S = Q @ K^T for this 16-key tile (K rows from LDS, stride 68: no bank
    // conflicts — lane mr reads row mr)
    f32x8 s;
#pragma unroll
    for (int r = 0; r < 8; ++r) s[r] = 0.f;
#pragma unroll
    for (int f = 0; f < 16; ++f) {
      const int kb = f * 4 + hi * 2;
      f32x2 bk = {Ks[cur][mr][kb], Ks[cur][mr][kb + 1]};
      s = wmma4(qf[f], bk, s);
    }
    const bool valid = (kv0 + mr) < KVLEN;
    // online softmax per row (rows r+8*hi, cols live in this 16-lane half)
#pragma unroll
    for (int r = 0; r < 8; ++r) {
      const float sv = valid ? s[r] * SCALE : -1e30f;
      float mx = sv;
      mx = fmaxf(mx, __shfl_xor(mx, 1, 32));
      mx = fmaxf(mx, __shfl_xor(mx, 2, 32));
      mx = fmaxf(mx, __shfl_xor(mx, 4, 32));
      mx = fmaxf(mx, __shfl_xor(mx, 8, 32));
      const float mnew = fmaxf(mrow[r], mx);
      const float p = valid ? __expf(sv - mnew) : 0.f;
      float ts = p;
      ts += __shfl_xor(ts, 1, 32);
      ts += __shfl_xor(ts, 2, 32);
      ts += __shfl_xor(ts, 4, 32);
      ts += __shfl_xor(ts, 8, 32);
      const float alpha = __expf(mrow[r] - mnew);
      srow[r] = srow[r] * alpha + ts;
      mrow[r] = mnew;
#pragma unroll
      for (int t = 0; t < 4; ++t) oacc[t][r] *= alpha;
      Ps[wv][r + 8 * hi][mr] = p;  // C-layout (row, col) -> LDS
    }
    asm volatile("s_wait_dscnt 0" ::: "memory");
    // reload P as A-fragments (m = mr, k = kv within tile)
    f32x2 pa[4];
#pragma unroll
    for (int f = 0; f < 4; ++f) {
      const int kk = f * 4 + hi * 2;
      pa[f][0] = Ps[wv][mr][kk];
      pa[f][1] = Ps[wv][mr][kk + 1];
    }
    // O += P @ V_tile (V rows from LDS)
#pragma unroll
    for (int t = 0; t < 4; ++t) {
      const int nn = t * 16 + mr;
#pragma unroll
      for (int f = 0; f < 4; ++f) {
        const int kb = f * 4 + hi * 2;
        f32x2 vb = {Vs[cur][kb][nn], Vs[cur][kb + 1][nn]};
        oacc[t] = wmma4(pa[f], vb, oacc[t]);
      }
    }
    __syncthreads();  // all waves done with buf[cur] before it is re-DMA'd
  }
  // epilogue: divide by softmax denom, write o in [B, L, H, HD] layout
#pragma unroll
  for (int r = 0; r < 8; ++r) {
    const float inv = 1.f / srow[r];
    const int row = q0 + r + 8 * hi;
    float* orow = o + ((size_t)(bq * SEQ + row) * NHEAD + h) * HDIM;
#pragma unroll
    for (int t = 0; t < 4; ++t) orow[t * 16 + mr] = oacc[t][r] * inv;
  }
}

// Output projection + residual: y = x + o @ out_w
__global__ __launch_bounds__(256) void out_proj_kernel(
    const float* __restrict__ o, const float* __restrict__ w,
    const float* __restrict__ x, float* __restrict__ y) {
  const int lane = threadIdx.x & 31;
  const int W = __builtin_amdgcn_readfirstlane(blockIdx.x * 8 +
                                               (threadIdx.x >> 5));
  const int TN = DMODEL / 64;  // 16
  const int m0 = (W / TN) * 32, n0 = (W % TN) * 64;
  const int hi = lane >> 4, mr = lane & 15;

  f32x8 acc[2][4];
#pragma unroll
  for (int i = 0; i < 2; ++i)
#pragma unroll
    for (int t = 0; t < 4; ++t)
#pragma unroll
      for (int r = 0; r < 8; ++r) acc[i][t][r] = 0.f;

  gemm32x64(o + (size_t)m0 * DMODEL, DMODEL, w + n0, DMODEL, DMODEL, lane, acc);

#pragma unroll
  for (int mi = 0; mi < 2; ++mi)
#pragma unroll
    for (int t = 0; t < 4; ++t) {
      const int col = n0 + t * 16 + mr;
#pragma unroll
      for (int r = 0; r < 8; ++r) {
        const size_t tok = m0 + mi * 16 + r + 8 * hi;
        y[tok * DMODEL + col] = x[tok * DMODEL + col] + acc[mi][t][r];
      }
    }
}

// ---------------------------------------------------------------------------
extern "C" void kernel_launch(void* const* d_in, const int* in_sizes, int n_in,
                              void* d_out, int out_size, void* d_ws,
                              size_t ws_size, hipStream_t stream) {
  const float* x     = (const float*)d_in[0];
  const float* qkv_w = (const float*)d_in[1];
  const float* out_w = (const float*)d_in[2];
  const float* ln_g  = (const float*)d_in[3];
  const float* ln_b  = (const float*)d_in[4];
  const float* pk0   = (const float*)d_in[5];
  const float* pv0   = (const float*)d_in[6];
  const float* kw1   = (const float*)d_in[7];
  const float* kb1   = (const float*)d_in[8];
  const float* kw2   = (const float*)d_in[9];
  const float* kb2   = (const float*)d_in[10];
  const float* vw1   = (const float*)d_in[11];
  const float* vb1   = (const float*)d_in[12];
  const float* vw2   = (const float*)d_in[13];
  const float* vb2   = (const float*)d_in[14];
  float* out = (float*)d_out;

  float* ws = (float*)d_ws;
  const size_t SZ_XN = (size_t)NTOK * DMODEL;                  // 8.39M
  const size_t SZ_Q  = (size_t)NBATCH * NHEAD * SEQ * HDIM;    // 8.39M
  const size_t SZ_KV = (size_t)NBATCH * NHEAD * KVPAD * HDIM;  // 8.52M
  float* xn   = ws;               ws += SZ_XN;
  float* q    = ws;               ws += SZ_Q;
  float* kbuf = ws;               ws += SZ_KV;
  float* vbuf = ws;               ws += SZ_KV;
  float* ob   = ws;               ws += SZ_XN;
  float* hk   = ws;               ws += NPFX * 512;
  float* pk   = ws;               ws += NPFX * DMODEL;
  float* hv   = ws;               ws += NPFX * 512;
  float* pv   = ws;               ws += NPFX * DMODEL;

  // 1. LayerNorm (wave per row)
  ln_kernel<<<NTOK / 8, 256, 0, stream>>>(x, ln_g, ln_b, xn);
  // 2. Prefix MLPs (tiny)
  mlp_kernel<<<(NPFX * 512 + 255) / 256, 256, 0, stream>>>(pk0, kw1, kb1, hk,
                                                           NPFX, DMODEL, 512, 1);
  mlp_kernel<<<(NPFX * DMODEL + 255) / 256, 256, 0, stream>>>(hk, kw2, kb2, pk,
                                                              NPFX, 512, DMODEL, 0);
  mlp_kernel<<<(NPFX * 512 + 255) / 256, 256, 0, stream>>>(pv0, vw1, vb1, hv,
                                                           NPFX, DMODEL, 512, 1);
  mlp_kernel<<<(NPFX * DMODEL + 255) / 256, 256, 0, stream>>>(hv, vw2, vb2, pv,
                                                              NPFX, 512, DMODEL, 0);
  // 3. Broadcast prefixes + zero kv pad rows
  prefix_fill_kernel<<<(NBATCH * NHEAD * 32 * HDIM + 255) / 256, 256, 0,
                       stream>>>(pk, pv, kbuf, vbuf);
  // 4. QKV projection (fp32 WMMA)
  qkv_gemm_kernel<<<(NTOK / 32) * (ND3 / 64) / 8, 256, 0, stream>>>(
      xn, qkv_w, q, kbuf, vbuf);
  // 5. Flash attention (fp32 WMMA + TDM-staged K/V tiles)
  attn_kernel<<<dim3(SEQ / 64, NHEAD, NBATCH), 128, 0, stream>>>(q, kbuf, vbuf,
                                                                 ob);
  // 6. Output projection + residual (fp32 WMMA)
  out_proj_kernel<<<(NTOK / 32) * (DMODEL / 64) / 8, 256, 0, stream>>>(
      ob, out_w, x, out);
}